// PrunedWeightedGroupSum_86277303042370
// MI455X (gfx1250) — compile-verified
//
#include <hip/hip_runtime.h>

typedef __attribute__((ext_vector_type(2))) float v2f;
typedef __attribute__((ext_vector_type(4))) float v4f;
typedef __attribute__((ext_vector_type(8))) float v8f;

#define IN_DIM   81920
#define NCLASS   1024
#define SPAN     1280            // features per 16-group k-tile
#define KT_STEPS (SPAN / 4)      // 320 K-steps of 4 per k-tile
#define HSTEPS   (KT_STEPS / 2)  // 160 steps per half-span (K-split of 2)

// ---------------------------------------------------------------------------
// Zero the output (two waves accumulate into each tile via fp atomics).
__global__ __launch_bounds__(256) void pwgs_zero(float* __restrict__ out) {
    size_t i = (size_t)blockIdx.x * 256 + threadIdx.x;   // 262144 v4f = 1M floats
    ((v4f*)out)[i] = (v4f){0.f, 0.f, 0.f, 0.f};
}

// ---------------------------------------------------------------------------
// Build the sparse-expanded B matrix directly in per-lane WMMA fragment
// layout: Bimg[((kt*320 + s)*32 + lane)*2 + {0,1}] holds B[K=hi*2+{0,1}][N=lane&15]
// for step s of k-tile kt.  wq = rint(max(w,0)) matches jnp.round (half-even).
// seg[f]&15 == seg[f]-16*kt inside tile kt (contiguous aligned groups).
__global__ __launch_bounds__(256) void pwgs_prep(const float* __restrict__ w,
                                                 const int*   __restrict__ seg,
                                                 float*       __restrict__ Bimg) {
    int t = blockIdx.x * 256 + threadIdx.x;              // 0 .. 655359
    int lane = t & 31;
    int gs   = t >> 5;                                   // global step 0..20479
    int kt   = gs / KT_STEPS;
    int sIn  = gs - kt * KT_STEPS;                       // step within k-tile
    int col  = lane & 15;
    int f0   = kt * SPAN + sIn * 4 + ((lane >> 4) << 1); // features f0, f0+1
    float b0 = ((seg[f0]     & 15) == col) ? rintf(fmaxf(w[f0],     0.f)) : 0.f;
    float b1 = ((seg[f0 + 1] & 15) == col) ? rintf(fmaxf(w[f0 + 1], 0.f)) : 0.f;
    v2f p; p.x = b0; p.y = b1;
    *(v2f*)(Bimg + 2 * (size_t)t) = p;
}

// ---------------------------------------------------------------------------
// One wave32 per 16x16 output tile per half-span.  8192 waves total.
// All 8 waves of a block share (kt, shalf) -> the 40KB B-image slice is
// WGP-cache resident; unique HBM traffic = x read exactly once (335 MB).
__global__ __launch_bounds__(256) void pwgs_wmma(const float* __restrict__ x,
                                                 const float* __restrict__ Bimg,
                                                 float*       __restrict__ out) {
    const int lane = threadIdx.x & 31;
    const int wib  = (threadIdx.x >> 5) & 7;    // wave in block
    const int b    = blockIdx.x;                // 0..1023
    const int kt    = b >> 4;                   // 0..63 : k-tile
    const int rem   = b & 15;
    const int shalf = rem >> 3;                 // 0..1  : half of the K span
    const int rt    = ((rem & 7) << 3) | wib;   // 0..63 : row-tile

    const int row = lane & 15;                  // A row == B column-in-tile
    const int hi  = lane >> 4;                  // 0: K=0,1   1: K=2,3

    const int fbase = kt * SPAN + shalf * (SPAN / 2);
    const float* xp = x + (size_t)(rt * 16 + row) * IN_DIM + fbase + hi * 2;
    const float* bp = Bimg
        + ((size_t)(kt * KT_STEPS + shalf * HSTEPS) * 32 + lane) * 2;

    v8f acc = {};
#pragma unroll 8
    for (int s = 0; s < HSTEPS; ++s) {
        v2f a  = *(const v2f*)(xp + 4 * s);     // 16x4 f32 A fragment
        v2f bf = *(const v2f*)(bp + 64 * s);    // 4x16 f32 B fragment
        acc = __builtin_amdgcn_wmma_f32_16x16x4_f32(
            /*neg_a=*/false, a, /*neg_b=*/false, bf,
            /*c_mod=*/(short)0, acc, /*reuse_a=*/false, /*reuse_b=*/false);
    }

    // C/D layout: VGPR v, lanes 0-15 -> M=v, lanes 16-31 -> M=v+8, N=lane&15.
    // Two half-span waves combine per element: fp add of 2 terms is exactly
    // commutative -> deterministic.
    float* op = out + (size_t)(rt * 16 + hi * 8) * NCLASS + kt * 16 + row;
#pragma unroll
    for (int v = 0; v < 8; ++v) {
        unsafeAtomicAdd(op + (size_t)v * NCLASS, acc[v]);
    }
}

// ---------------------------------------------------------------------------
extern "C" void kernel_launch(void* const* d_in, const int* in_sizes, int n_in,
                              void* d_out, int out_size, void* d_ws, size_t ws_size,
                              hipStream_t stream) {
    const float* x   = (const float*)d_in[0];   // [1024, 81920] f32
    const float* w   = (const float*)d_in[1];   // [81920] f32
    const int*   seg = (const int*)d_in[2];     // [81920] i32
    float* Bimg = (float*)d_ws;                 // 64*320*32*2 f32 = 5.24 MB
    float* out  = (float*)d_out;                // [1024, 1024] f32

    pwgs_zero<<<1024, 256, 0, stream>>>(out);
    pwgs_prep<<<(64 * KT_STEPS * 32) / 256, 256, 0, stream>>>(w, seg, Bimg);
    // 64 k-tiles * 2 half-spans * 64 row-tiles = 8192 waves; 8 waves/block.
    pwgs_wmma<<<1024, 256, 0, stream>>>(x, Bimg, out);
}